// DeformableBlock_27822798143530
// MI455X (gfx1250) — compile-verified
//
#include <hip/hip_runtime.h>
#include <cmath>

typedef float v2f __attribute__((ext_vector_type(2)));
typedef float v8f __attribute__((ext_vector_type(8)));

#define HH   128
#define WWD  128
#define HW   (128*128)
#define CIN  64
#define COUT 64
#define KTOT 576   // 9 taps * 64 channels

// ---------------------------------------------------------------------------
// K1: 3x3 offset conv (64 -> 18 ch), writes dy/dx planes to ws.
// ---------------------------------------------------------------------------
__global__ __launch_bounds__(256) void k_offsets(const float* __restrict__ x,
                                                 const float* __restrict__ w_off,
                                                 const float* __restrict__ b_off,
                                                 float* __restrict__ dydx) {
  __shared__ float wsm[18 * 64 * 9];
  for (int d = threadIdx.x; d < 18 * 64 * 9; d += 256) wsm[d] = w_off[d];
  __syncthreads();

  int tid = blockIdx.x * 256 + threadIdx.x;   // 0 .. 65535
  int b = tid >> 14;
  int p = tid & 16383;
  int h = p >> 7, w = p & 127;

  float acc[18];
#pragma unroll
  for (int o = 0; o < 18; ++o) acc[o] = b_off[o];

  const float* xb = x + (size_t)b * CIN * HW;
  for (int c = 0; c < CIN; ++c) {
    const float* xc = xb + c * HW;
    float xv[9];
#pragma unroll
    for (int t = 0; t < 9; ++t) {
      int yy = h + t / 3 - 1, xx = w + t % 3 - 1;
      bool ok = (yy >= 0) & (yy < HH) & (xx >= 0) & (xx < WWD);
      xv[t] = ok ? xc[yy * WWD + xx] : 0.f;
    }
#pragma unroll
    for (int o = 0; o < 18; ++o) {
      const float* wr = &wsm[(o * 64 + c) * 9];
#pragma unroll
      for (int t = 0; t < 9; ++t) acc[o] = fmaf(xv[t], wr[t], acc[o]);
    }
  }
#pragma unroll
  for (int k = 0; k < 9; ++k) {
    dydx[(size_t)((b * 9 + k) * 2 + 0) * HW + p] = acc[2 * k];
    dydx[(size_t)((b * 9 + k) * 2 + 1) * HW + p] = acc[2 * k + 1];
  }
}

// ---------------------------------------------------------------------------
// K2: fused bilinear gather + deformable GEMM via V_WMMA_F32_16X16X4_F32.
// Block = 256 threads = 8 waves; block covers one image row (128 pixels) of
// one batch. Wave wv owns pixels [16*wv, 16*wv+16) x all 64 outputs.
// K' ordering: K' = tap*64 + c  (bilinear params reused across 16 chunks).
// ---------------------------------------------------------------------------
__global__ __launch_bounds__(256) void k_deform_gemm(const float* __restrict__ x,
                                                     const float* __restrict__ w_def,
                                                     const float* __restrict__ b_def,
                                                     const float* __restrict__ dydx,
                                                     float* __restrict__ out) {
  // One tap worth of B, pre-swizzled to fragment layout:
  // Bsm[((cl*4 + nt)*32 + lane)*2 + j] = w_def[n*576 + c*9 + tap]
  //   with c = cl*4 + 2*(lane>=16) + j, n = nt*16 + (lane&15)
  __shared__ float Bsm[4096];

  int blk = blockIdx.x;          // 0..511
  int b = blk >> 7;
  int h = blk & 127;
  int t = threadIdx.x;
  int lane = t & 31;
  int wv = t >> 5;
  int hi = lane >> 4;            // 0: lanes 0-15, 1: lanes 16-31
  int m  = lane & 15;
  int w  = wv * 16 + m;          // pixel column this lane gathers for

  v8f acc0 = {}, acc1 = {}, acc2 = {}, acc3 = {};

  for (int tap = 0; tap < 9; ++tap) {
    __syncthreads();             // protect previous tap's Bsm reads
#pragma unroll
    for (int i = 0; i < 16; ++i) {
      int d  = t + i * 256;
      int j  = d & 1;
      int ln = (d >> 1) & 31;
      int nt = (d >> 6) & 3;
      int cl = (d >> 8) & 15;
      int c  = cl * 4 + ((ln >> 4) << 1) + j;
      int n  = nt * 16 + (ln & 15);
      Bsm[d] = w_def[n * KTOT + c * 9 + tap];
    }
    __syncthreads();

    // Bilinear parameters for (pixel w, tap): registers, reused 16 chunks.
    float dy = dydx[(size_t)((b * 9 + tap) * 2 + 0) * HW + h * WWD + w];
    float dx = dydx[(size_t)((b * 9 + tap) * 2 + 1) * HW + h * WWD + w];
    float py = (float)h + (float)(tap / 3 - 1) + dy;
    float px = (float)w + (float)(tap % 3 - 1) + dx;
    float y0f = floorf(py), x0f = floorf(px);
    float wy = py - y0f, wx = px - x0f;
    float vy0 = (y0f >= 0.f  && y0f <= 127.f) ? 1.f : 0.f;
    float vy1 = (y0f >= -1.f && y0f <= 126.f) ? 1.f : 0.f;
    float vx0 = (x0f >= 0.f  && x0f <= 127.f) ? 1.f : 0.f;
    float vx1 = (x0f >= -1.f && x0f <= 126.f) ? 1.f : 0.f;
    int iy0 = min(max((int)y0f, 0), 127);
    int iy1 = min(max((int)y0f + 1, 0), 127);
    int ix0 = min(max((int)x0f, 0), 127);
    int ix1 = min(max((int)x0f + 1, 0), 127);
    float w00 = (1.f - wy) * (1.f - wx) * vy0 * vx0;
    float w01 = (1.f - wy) * wx * vy0 * vx1;
    float w10 = wy * (1.f - wx) * vy1 * vx0;
    float w11 = wy * wx * vy1 * vx1;
    int i00 = iy0 * WWD + ix0, i01 = iy0 * WWD + ix1;
    int i10 = iy1 * WWD + ix0, i11 = iy1 * WWD + ix1;

    const float* xb = x + (size_t)(b * CIN) * HW;

    for (int c0 = 0; c0 < 64; c0 += 4) {
      int ca = c0 + (hi << 1);
      const float* xc0 = xb + ca * HW;
      const float* xc1 = xc0 + HW;
      v2f a;
      a.x = w00 * xc0[i00] + w01 * xc0[i01] + w10 * xc0[i10] + w11 * xc0[i11];
      a.y = w00 * xc1[i00] + w01 * xc1[i01] + w10 * xc1[i10] + w11 * xc1[i11];

      int cl = c0 >> 2;
      v2f b0 = *(const v2f*)&Bsm[((cl * 4 + 0) * 32 + lane) * 2];
      v2f b1 = *(const v2f*)&Bsm[((cl * 4 + 1) * 32 + lane) * 2];
      v2f b2 = *(const v2f*)&Bsm[((cl * 4 + 2) * 32 + lane) * 2];
      v2f b3 = *(const v2f*)&Bsm[((cl * 4 + 3) * 32 + lane) * 2];

      acc0 = __builtin_amdgcn_wmma_f32_16x16x4_f32(false, a, false, b0, (short)0, acc0, false, false);
      acc1 = __builtin_amdgcn_wmma_f32_16x16x4_f32(false, a, false, b1, (short)0, acc1, false, false);
      acc2 = __builtin_amdgcn_wmma_f32_16x16x4_f32(false, a, false, b2, (short)0, acc2, false, false);
      acc3 = __builtin_amdgcn_wmma_f32_16x16x4_f32(false, a, false, b3, (short)0, acc3, false, false);
    }
  }

  // Epilogue: D layout — VGPR r: M = r + 8*hi, N = lane&15. Add conv bias.
#pragma unroll
  for (int r = 0; r < 8; ++r) {
    int pm  = wv * 16 + r + 8 * hi;
    int row = h * WWD + pm;
    out[(size_t)(b * COUT + ( 0 + m)) * HW + row] = acc0[r] + b_def[ 0 + m];
    out[(size_t)(b * COUT + (16 + m)) * HW + row] = acc1[r] + b_def[16 + m];
    out[(size_t)(b * COUT + (32 + m)) * HW + row] = acc2[r] + b_def[32 + m];
    out[(size_t)(b * COUT + (48 + m)) * HW + row] = acc3[r] + b_def[48 + m];
  }
}

// ---------------------------------------------------------------------------
// K3: GroupNorm statistics. One block per (b, group): 4 contiguous channels.
// ---------------------------------------------------------------------------
__global__ __launch_bounds__(256) void k_gn_stats(const float* __restrict__ out,
                                                  float* __restrict__ stats) {
  int bg = blockIdx.x;           // 0..63
  int b = bg >> 4, g = bg & 15;
  const float* base = out + (size_t)(b * COUT + g * 4) * HW;
  float s = 0.f, s2 = 0.f;
  for (int i = threadIdx.x; i < 4 * HW; i += 256) {
    float v = base[i];
    s += v; s2 += v * v;
  }
  __shared__ float rs[256], rq[256];
  rs[threadIdx.x] = s; rq[threadIdx.x] = s2;
  __syncthreads();
  for (int off = 128; off > 0; off >>= 1) {
    if (threadIdx.x < off) {
      rs[threadIdx.x] += rs[threadIdx.x + off];
      rq[threadIdx.x] += rq[threadIdx.x + off];
    }
    __syncthreads();
  }
  if (threadIdx.x == 0) {
    float mean = rs[0] * (1.f / 65536.f);
    float var  = rq[0] * (1.f / 65536.f) - mean * mean;
    stats[bg * 2 + 0] = mean;
    stats[bg * 2 + 1] = rsqrtf(var + 1e-5f);
  }
}

// ---------------------------------------------------------------------------
// K4: normalize + affine + LeakyReLU, in place on d_out.
// ---------------------------------------------------------------------------
__global__ __launch_bounds__(256) void k_gn_apply(float* __restrict__ out,
                                                  const float* __restrict__ stats,
                                                  const float* __restrict__ gn_w,
                                                  const float* __restrict__ gn_b) {
  int i = blockIdx.x * 256 + threadIdx.x;   // 0 .. 4*64*HW-1
  int b = i >> 20;                          // 64*HW = 2^20 per batch
  int o = (i >> 14) & 63;
  int g = o >> 2;
  float mean = stats[(b * 16 + g) * 2 + 0];
  float rstd = stats[(b * 16 + g) * 2 + 1];
  float v = out[i];
  float y = (v - mean) * rstd * gn_w[o] + gn_b[o];
  out[i] = (y >= 0.f) ? y : 0.2f * y;
}

// ---------------------------------------------------------------------------
extern "C" void kernel_launch(void* const* d_in, const int* in_sizes, int n_in,
                              void* d_out, int out_size, void* d_ws, size_t ws_size,
                              hipStream_t stream) {
  const float* x     = (const float*)d_in[0];
  const float* w_off = (const float*)d_in[1];
  const float* b_off = (const float*)d_in[2];
  const float* w_def = (const float*)d_in[3];
  const float* b_def = (const float*)d_in[4];
  const float* gn_w  = (const float*)d_in[5];
  const float* gn_b  = (const float*)d_in[6];
  float* out = (float*)d_out;

  float* wsf   = (float*)d_ws;
  float* dydx  = wsf;                       // 4*9*2*HW floats = 4.7 MB
  float* stats = wsf + (size_t)4 * 9 * 2 * HW;  // 128 floats

  k_offsets    <<<256,   256, 0, stream>>>(x, w_off, b_off, dydx);
  k_deform_gemm<<<512,   256, 0, stream>>>(x, w_def, b_def, dydx, out);
  k_gn_stats   <<<64,    256, 0, stream>>>(out, stats);
  k_gn_apply   <<<16384, 256, 0, stream>>>(out, stats, gn_w, gn_b);
}